// SparseLinear_49538152792604
// MI455X (gfx1250) — compile-verified
//
#include <hip/hip_runtime.h>
#include <stdint.h>

// Problem constants (match reference)
#define IN_FEATURES  16384
#define OUT_FEATURES 16384
#define NNZ          1000000
#define BATCH        256

typedef __attribute__((ext_vector_type(2))) float v2f;
typedef __attribute__((ext_vector_type(8))) float v8f;

// ---------------------------------------------------------------------------
// Kernel 1: outT[r][b] = bias[r]  (also clears workspace poison).
// The bias broadcast is the rank-1 outer product  outT_tile = bias_seg . 1^T,
// computed exactly on the matrix pipe with V_WMMA_F32_16X16X4_F32:
//   A[16x4] = [bias | 0 0 0],  B[4x16] = [1...1 ; 0 ; 0 ; 0]  =>  D[m][n]=bias[m]
// One wave per 16(row) x 16(batch) tile; 16384*256/256 = 16384 waves.
// A layout (16x4 f32): lane 0-15 = M, VGPR0={K0,K2}, VGPR1={K1,K3}.
// B layout (4x16 f32): VGPR0 = K0 row (lanes 0-15) / K2 row (lanes 16-31).
// D layout: VGPR v = row M=v (lanes 0-15, N=lane) / M=v+8 (lanes 16-31).
// ---------------------------------------------------------------------------
__global__ void sl_init_outT_wmma(const float* __restrict__ bias,
                                  float* __restrict__ outT) {
    const int lane = threadIdx.x & 31;
    const int wave = (blockIdx.x * blockDim.x + threadIdx.x) >> 5;
    const int r0 = (wave >> 4) << 4;   // 16-row strip
    const int b0 = (wave & 15) << 4;   // 16-batch-col strip
    const int half = lane >> 4;        // 0: lanes 0-15, 1: lanes 16-31
    const int l15  = lane & 15;

    const float bv = bias[r0 + l15];   // uniform-shape load, selected below

    v2f a;                              // A[m][k]
    a.x = (half == 0) ? bv : 0.0f;      // K=0 column = bias (K=2 half -> 0)
    a.y = 0.0f;                         // K=1 / K=3 columns = 0
    v2f b;                              // B[k][n]
    b.x = (half == 0) ? 1.0f : 0.0f;    // K=0 row = ones (K=2 row -> 0)
    b.y = 0.0f;                         // K=1 / K=3 rows = 0
    v8f c = {};
    // D = A x B + 0  (f32 x f32 -> f32, exact)
    v8f d = __builtin_amdgcn_wmma_f32_16x16x4_f32(
        false, a, false, b, (short)0, c, false, false);

#pragma unroll
    for (int v = 0; v < 8; ++v) {
        const int m = v + half * 8;
        outT[(size_t)(r0 + m) * BATCH + b0 + l15] = d[v];
    }
}

// ---------------------------------------------------------------------------
// Kernel 2/4: 32x32 tiled transpose, dst[c][r] = src[r][c].
// Tile is staged in LDS via CDNA5 async global->LDS loads (ASYNCcnt),
// then written out coalesced. LDS row stride 33 -> conflict-free reads.
// Block = 256 threads (8 wave32s), each thread moves 4 elements.
// ---------------------------------------------------------------------------
__global__ void sl_transpose32(const float* __restrict__ src,
                               float* __restrict__ dst,
                               int srcRows, int srcCols) {
    __shared__ float tile[32][33];
    const int c0 = blockIdx.x * 32;
    const int r0 = blockIdx.y * 32;
    const int tx = threadIdx.x & 31;   // column within tile (coalesced dim)
    const int ty = threadIdx.x >> 5;   // 0..7

#pragma unroll
    for (int k = 0; k < 4; ++k) {
        const int r = ty + 8 * k;
        const float* g = src + (size_t)(r0 + r) * srcCols + (c0 + tx);
        // LDS byte offset = low 32 bits of the flat shared-aperture address.
        unsigned lds_off = (unsigned)(uintptr_t)(&tile[r][tx]);
        unsigned long long gaddr = (unsigned long long)(uintptr_t)g;
        // CDNA5 async load: copies one global dword into LDS at lds_off, tracked by ASYNCcnt.
        asm volatile("global_load_async_to_lds_b32 %0, %1, off"
                     :: "v"(lds_off), "v"(gaddr)
                     : "memory");
    }
    // Drain this wave's async copies, then make LDS visible across waves.
    asm volatile("s_wait_asynccnt 0x0" ::: "memory");
    __syncthreads();

#pragma unroll
    for (int k = 0; k < 4; ++k) {
        const int c = ty + 8 * k;
        dst[(size_t)(c0 + c) * srcRows + (r0 + tx)] = tile[tx][c];
    }
}

// ---------------------------------------------------------------------------
// Kernel 3: edge scatter. One wave32 per edge iteration:
//   lane t handles batch b = t + 32k, k = 0..7  (each instruction's 32 lanes
//   cover one contiguous 128B run -> fully coalesced loads & atomics).
//   outT[r][b] += v * xT[c][b]  via f32 L2 atomics (outT is L2-resident).
// ---------------------------------------------------------------------------
__global__ void sl_scatter(const float* __restrict__ xT,
                           const float* __restrict__ vals,
                           const int*   __restrict__ rows,
                           const int*   __restrict__ cols,
                           float*       __restrict__ outT,
                           int nnz) {
    const int lane   = threadIdx.x & 31;
    const int gwave  = (blockIdx.x * blockDim.x + threadIdx.x) >> 5;
    const int nwaves = (gridDim.x * blockDim.x) >> 5;

    for (int e = gwave; e < nnz; e += nwaves) {
        // Hint the next edge batch into cache (gfx1250 global_prefetch_b8).
        if (e + nwaves < nnz) {
            __builtin_prefetch(&rows[e + nwaves], 0, 1);
            __builtin_prefetch(&cols[e + nwaves], 0, 1);
            __builtin_prefetch(&vals[e + nwaves], 0, 1);
        }
        const int   r = rows[e];
        const int   c = cols[e];
        const float v = vals[e];
        const float* xs = xT  + (size_t)c * BATCH;
        float*       os = outT + (size_t)r * BATCH;
#pragma unroll
        for (int k = 0; k < 8; ++k) {
            const int b = lane + 32 * k;
            unsafeAtomicAdd(&os[b], v * xs[b]);   // global_atomic_add_f32
        }
    }
}

// ---------------------------------------------------------------------------
// Launch: init outT=bias (WMMA outer product) -> transpose x->xT ->
//         atomic scatter -> transpose outT->out
// Workspace: xT (16 MB) | outT (16 MB)  = 32 MB of d_ws.
// ---------------------------------------------------------------------------
extern "C" void kernel_launch(void* const* d_in, const int* in_sizes, int n_in,
                              void* d_out, int out_size, void* d_ws, size_t ws_size,
                              hipStream_t stream) {
    (void)in_sizes; (void)n_in; (void)out_size; (void)ws_size;

    const float* x    = (const float*)d_in[0];   // [BATCH, IN_FEATURES]
    const float* wval = (const float*)d_in[1];   // [NNZ]
    const float* bias = (const float*)d_in[2];   // [OUT_FEATURES]
    const int*   rows = (const int*)  d_in[3];   // [NNZ]
    const int*   cols = (const int*)  d_in[4];   // [NNZ]
    float*       out  = (float*)d_out;           // [BATCH, OUT_FEATURES]

    float* xT   = (float*)d_ws;                          // [IN_FEATURES][BATCH]
    float* outT = xT + (size_t)IN_FEATURES * BATCH;      // [OUT_FEATURES][BATCH]

    // 1) outT = bias . 1^T via V_WMMA_F32_16X16X4_F32 (16384 tile-waves)
    sl_init_outT_wmma<<<(OUT_FEATURES / 16) * (BATCH / 16) / 8, 256, 0, stream>>>(
        bias, outT);

    // 2) xT = transpose(x)   src is [BATCH, IN_FEATURES]
    sl_transpose32<<<dim3(IN_FEATURES / 32, BATCH / 32), 256, 0, stream>>>(
        x, xT, BATCH, IN_FEATURES);

    // 3) atomic edge scatter (32768 waves, ~31 edges each)
    sl_scatter<<<4096, 256, 0, stream>>>(xT, wval, rows, cols, outT, NNZ);

    // 4) out = transpose(outT)   src is [OUT_FEATURES, BATCH]
    sl_transpose32<<<dim3(BATCH / 32, OUT_FEATURES / 32), 256, 0, stream>>>(
        outT, out, OUT_FEATURES, BATCH);
}